// PtrNet_40475771797721
// MI455X (gfx1250) — compile-verified
//
#include <hip/hip_runtime.h>
#include <hip/hip_bf16.h>

// PtrNet on MI455X (gfx1250). fp32 WMMA (V_WMMA_F32_16X16X4_F32) since the
// workload is weight-streaming bandwidth bound (~820MB unique weights/call,
// ~25 GFLOP -> ~30 FLOP/byte at fp32; HBM 23.3TB/s is the wall).

#define PN_B 64
#define PN_S 50
#define PN_H 512
#define PN_FH 2048

typedef float v2f __attribute__((ext_vector_type(2)));
typedef float v8f __attribute__((ext_vector_type(8)));

__device__ __forceinline__ float pn_sig(float x) { return 1.0f / (1.0f + expf(-x)); }

// One 16x16 fp32 WMMA tile, K=512 reduction.
// Arow: points at A row (m = lane&15) + 2*(lane>>4) floats in; stride PN_H.
// Brow: points at W row (n = lane&15) + 2*(lane>>4) floats in; stride PN_H.
// D[m][n] = sum_k A[m][k] * W[n][k]   (i.e. A @ W.T tile)
__device__ __forceinline__ v8f pn_gemm_k512(const float* __restrict__ Arow,
                                            const float* __restrict__ Brow,
                                            v8f acc) {
#pragma unroll 8
  for (int k = 0; k < PN_H; k += 4) {
    v2f a = *(const v2f*)(Arow + k);
    v2f b = *(const v2f*)(Brow + k);
    acc = __builtin_amdgcn_wmma_f32_16x16x4_f32(false, a, false, b, (short)0,
                                                acc, false, false);
  }
  return acc;
}

__global__ void pn_init_state(float* __restrict__ z0, float* __restrict__ c,
                              float* __restrict__ d) {
  int i = blockIdx.x * 256 + threadIdx.x;
  z0[i] = 0.0f;
  c[i] = 0.0f;
  d[i] = -1.0f;
}

// Encoder step: gates = x_t outer Wih + h @ Whh.T + bih + bhh ; LSTM cell.
// Grid: 128 blocks (32 n-tiles x 4 m-tiles), 128 threads (4 waves = 4 gates).
__global__ void pn_enc_step(const float* __restrict__ hin,
                            const float* __restrict__ x,
                            const float* __restrict__ Wih,
                            const float* __restrict__ Whh,
                            const float* __restrict__ bih,
                            const float* __restrict__ bhh,
                            float* __restrict__ c, float* __restrict__ henc,
                            int s) {
  __shared__ float lds[4][16][16];
  const int tid = threadIdx.x;
  const int g = tid >> 5;  // wave index = gate (i,f,g,o)
  const int lane = tid & 31;
  const int half = lane >> 4;
  const int ln = lane & 15;
  const int nt = blockIdx.x & 31;
  const int mt = blockIdx.x >> 5;
  const int M0 = mt * 16;
  const int j0 = g * PN_H + nt * 16;

  const float* Arow = hin + (size_t)(M0 + ln) * PN_H + 2 * half;
  const float* Brow = Whh + (size_t)(j0 + ln) * PN_H + 2 * half;
  v8f acc = {};
  acc = pn_gemm_k512(Arow, Brow, acc);

  const int j = j0 + ln;
  const float wih = Wih[j];
  const float bb = bih[j] + bhh[j];
#pragma unroll
  for (int r = 0; r < 8; ++r) {
    int m = r + 8 * half;  // local batch row 0..15
    lds[g][m][ln] = acc[r] + x[(size_t)(M0 + m) * PN_S + s] * wih + bb;
  }
  __syncthreads();

  for (int e0 = tid; e0 < 256; e0 += 128) {
    int bl = e0 >> 4;
    int t = e0 & 15;
    int row = M0 + bl;
    int hcol = nt * 16 + t;
    float gi = lds[0][bl][t], gf = lds[1][bl][t];
    float gg = lds[2][bl][t], go = lds[3][bl][t];
    size_t idx = (size_t)row * PN_H + hcol;
    float c2 = pn_sig(gf) * c[idx] + pn_sig(gi) * tanhf(gg);
    float h2 = pn_sig(go) * tanhf(c2);
    c[idx] = c2;
    henc[idx] = h2;
  }
}

// Decoder gates: gates = d @ Wih.T + h @ Whh.T + bih + bhh ; LSTM cell.
__global__ void pn_dec_gates(const float* __restrict__ hin,
                             const float* __restrict__ din,
                             const float* __restrict__ Wih,
                             const float* __restrict__ Whh,
                             const float* __restrict__ bih,
                             const float* __restrict__ bhh,
                             float* __restrict__ c, float* __restrict__ hout) {
  __shared__ float lds[4][16][16];
  const int tid = threadIdx.x;
  const int g = tid >> 5;
  const int lane = tid & 31;
  const int half = lane >> 4;
  const int ln = lane & 15;
  const int nt = blockIdx.x & 31;
  const int mt = blockIdx.x >> 5;
  const int M0 = mt * 16;
  const int j0 = g * PN_H + nt * 16;

  v8f acc = {};
  acc = pn_gemm_k512(din + (size_t)(M0 + ln) * PN_H + 2 * half,
                     Wih + (size_t)(j0 + ln) * PN_H + 2 * half, acc);
  acc = pn_gemm_k512(hin + (size_t)(M0 + ln) * PN_H + 2 * half,
                     Whh + (size_t)(j0 + ln) * PN_H + 2 * half, acc);

  const int j = j0 + ln;
  const float bb = bih[j] + bhh[j];
#pragma unroll
  for (int r = 0; r < 8; ++r) {
    int m = r + 8 * half;
    lds[g][m][ln] = acc[r] + bb;
  }
  __syncthreads();

  for (int e0 = tid; e0 < 256; e0 += 128) {
    int bl = e0 >> 4;
    int t = e0 & 15;
    int row = M0 + bl;
    int hcol = nt * 16 + t;
    float gi = lds[0][bl][t], gf = lds[1][bl][t];
    float gg = lds[2][bl][t], go = lds[3][bl][t];
    size_t idx = (size_t)row * PN_H + hcol;
    float c2 = pn_sig(gf) * c[idx] + pn_sig(gi) * tanhf(gg);
    float h2 = pn_sig(go) * tanhf(c2);
    c[idx] = c2;
    hout[idx] = h2;
  }
}

// Generic out[m,n] = A[m,:] . W[n,:] + bias[n].  8 waves/block, 1 tile/wave.
// tiles laid out mt-major over N tiles (PN_H/16 = 32 of them).
__global__ void pn_proj(const float* __restrict__ A, const float* __restrict__ W,
                        const float* __restrict__ bias, float* __restrict__ out) {
  const int wave = threadIdx.x >> 5;
  const int lane = threadIdx.x & 31;
  const int half = lane >> 4;
  const int ln = lane & 15;
  const int tile = blockIdx.x * 8 + wave;
  const int mt = tile >> 5;        // / (PN_H/16)
  const int nt = tile & 31;

  v8f acc = {};
  acc = pn_gemm_k512(A + (size_t)(mt * 16 + ln) * PN_H + 2 * half,
                     W + (size_t)(nt * 16 + ln) * PN_H + 2 * half, acc);

  const int n = nt * 16 + ln;
  const float bs = bias[n];
#pragma unroll
  for (int r = 0; r < 8; ++r) {
    int m = mt * 16 + r + 8 * half;
    out[(size_t)m * PN_H + n] = acc[r] + bs;
  }
}

// u[s,b] = V_w . tanh(ci[b,:] + e[s,b,:]) + V_b.  One wave per (s,b).
__global__ void pn_attn_u(const float* __restrict__ ci, const float* __restrict__ e,
                          const float* __restrict__ Vw, const float* __restrict__ Vb,
                          float* __restrict__ u) {
  const int wid = blockIdx.x * 8 + (threadIdx.x >> 5);
  const int lane = threadIdx.x & 31;
  const int s = wid / PN_B;
  const int b = wid % PN_B;
  const float* ev = e + ((size_t)s * PN_B + b) * PN_H;
  const float* cv = ci + (size_t)b * PN_H;
  float sum = 0.0f;
  for (int h = lane; h < PN_H; h += 32) sum += Vw[h] * tanhf(cv[h] + ev[h]);
#pragma unroll
  for (int off = 16; off; off >>= 1) sum += __shfl_xor(sum, off, 32);
  if (lane == 0) u[(size_t)s * PN_B + b] = sum + Vb[0];
}

// Softmax over S per batch row, emit distribution, and context d2 = a . enc.
__global__ void pn_attn_sm(const float* __restrict__ u, const float* __restrict__ enc,
                           float* __restrict__ aout, float* __restrict__ dout) {
  __shared__ float av[PN_S];
  const int b = blockIdx.x;
  const int tid = threadIdx.x;
  if (tid < PN_S) av[tid] = u[(size_t)tid * PN_B + b];
  __syncthreads();
  if (tid == 0) {
    float mx = av[0];
    for (int i = 1; i < PN_S; ++i) mx = fmaxf(mx, av[i]);
    float ssum = 0.0f;
    for (int i = 0; i < PN_S; ++i) { float t = expf(av[i] - mx); av[i] = t; ssum += t; }
    float inv = 1.0f / ssum;
    for (int i = 0; i < PN_S; ++i) av[i] *= inv;
  }
  __syncthreads();
  if (tid < PN_S) aout[(size_t)b * PN_S + tid] = av[tid];
  for (int h = tid; h < PN_H; h += 256) {
    float sum = 0.0f;
    for (int s2 = 0; s2 < PN_S; ++s2)
      sum += av[s2] * enc[((size_t)s2 * PN_B + b) * PN_H + h];
    dout[(size_t)b * PN_H + h] = sum;
  }
}

extern "C" void kernel_launch(void* const* d_in, const int* in_sizes, int n_in,
                              void* d_out, int out_size, void* d_ws, size_t ws_size,
                              hipStream_t stream) {
  (void)in_sizes; (void)n_in; (void)out_size; (void)ws_size;
  const float* x    = (const float*)d_in[0];
  const float* eWih = (const float*)d_in[1];
  const float* eWhh = (const float*)d_in[2];
  const float* ebih = (const float*)d_in[3];
  const float* ebhh = (const float*)d_in[4];
  const float* dWih = (const float*)d_in[5];
  const float* dWhh = (const float*)d_in[6];
  const float* dbih = (const float*)d_in[7];
  const float* dbhh = (const float*)d_in[8];
  const float* Wew  = (const float*)d_in[9];
  const float* Web  = (const float*)d_in[10];
  const float* Wdw  = (const float*)d_in[11];
  const float* Wdb  = (const float*)d_in[12];
  const float* Vw   = (const float*)d_in[13];
  const float* Vb   = (const float*)d_in[14];
  float* out = (float*)d_out;

  float* ws = (float*)d_ws;
  const size_t SBH = (size_t)PN_S * PN_B * PN_H;
  const size_t BH  = (size_t)PN_B * PN_H;
  float* enc = ws;
  float* e   = enc + SBH;
  float* z0  = e + SBH;
  float* c   = z0 + BH;
  float* hA  = c + BH;
  float* hB  = hA + BH;
  float* dv  = hB + BH;
  float* ci  = dv + BH;
  float* u   = ci + BH;

  pn_init_state<<<(int)(BH / 256), 256, 0, stream>>>(z0, c, dv);

  // ---- encoder scan (sequential over S) ----
  for (int s = 0; s < PN_S; ++s) {
    const float* hin = s ? enc + (size_t)(s - 1) * BH : z0;
    pn_enc_step<<<128, 128, 0, stream>>>(
        hin, x, eWih + (size_t)s * PN_FH, eWhh + (size_t)s * PN_FH * PN_H,
        ebih + (size_t)s * PN_FH, ebhh + (size_t)s * PN_FH, c,
        enc + (size_t)s * BH, s);
  }

  // ---- e = enc @ We_w.T + We_b (M = S*B = 3200 rows -> 6400 tiles) ----
  pn_proj<<<800, 256, 0, stream>>>(enc, Wew, Web, e);

  // ---- decoder scan ----
  const float* hcur = enc + (size_t)(PN_S - 1) * BH;
  for (int s = 0; s < PN_S; ++s) {
    float* hout = (s & 1) ? hB : hA;
    pn_dec_gates<<<128, 128, 0, stream>>>(
        hcur, dv, dWih + (size_t)s * PN_FH * PN_H,
        dWhh + (size_t)s * PN_FH * PN_H, dbih + (size_t)s * PN_FH,
        dbhh + (size_t)s * PN_FH, c, hout);
    hcur = hout;
    // ci = c2 @ Wd_w.T + Wd_b  (M = 64 -> 128 tiles)
    pn_proj<<<16, 256, 0, stream>>>(c, Wdw, Wdb, ci);
    pn_attn_u<<<400, 256, 0, stream>>>(ci, e, Vw, Vb, u);
    pn_attn_sm<<<PN_B, 256, 0, stream>>>(u, enc,
                                         out + (size_t)s * PN_B * PN_S, dv);
  }
}